// Deform_7499012899334
// MI455X (gfx1250) — compile-verified
//
#include <hip/hip_runtime.h>
#include <stdint.h>

// Deformable-warp bilinear grid sample, MI455X (gfx1250).
//
// Memory-bound kernel: ~184.5 MB NT-stored output + 11.5 MB motion reads;
// the 2 MB source image stays resident in the 192 MB L2 and feeds the
// gathers. HBM floor ~8.4 us @ 23.3 TB/s. WMMA does not apply (per-pixel
// gather operands, no shared matrix), so the CDNA5-specific path used here
// is the async global->LDS engine (ASYNCcnt) for staging motion coords.

typedef float v4f __attribute__((ext_vector_type(4)));

#define HH 128
#define WW 128
#define CC 32
#define NPIX (8 * 11 * 128 * 128)   // 1,441,792 pixels
#define TPB 256                     // 8 waves (wave32)
#define PIX_PER_BLK 32              // 8 lanes per pixel (8 x float4 = 32 ch)

__global__ __launch_bounds__(TPB) void deform_grid_sample_kernel(
    const float* __restrict__ src,      // (128,128,32)
    const float* __restrict__ motions,  // (NPIX, 2)
    float* __restrict__ out)            // (NPIX, 32)
{
    __shared__ float sm_motion[PIX_PER_BLK * 2];

    const int t    = threadIdx.x;
    const int base = blockIdx.x * PIX_PER_BLK;

    // --- Async-stage this block's 32 motion float2's into LDS (256 B). ---
    // One 8-byte async transfer per pixel; dedups the 8x redundant reads
    // the 8 lanes-per-pixel would otherwise issue.
    if (t < PIX_PER_BLK) {
        uint64_t ga = (uint64_t)(uintptr_t)(motions + (size_t)(base + t) * 2);
        uint32_t la = (uint32_t)(uintptr_t)(&sm_motion[t * 2]); // low 32b == LDS offset
        asm volatile("global_load_async_to_lds_b64 %0, %1, off"
                     :: "v"(la), "v"(ga) : "memory");
    }
    asm volatile("s_wait_asynccnt 0x0" ::: "memory");
    __syncthreads();

    const int lp    = t >> 3;   // local pixel 0..31
    const int chunk = t & 7;    // float4 chunk 0..7 (channels chunk*4 .. +3)
    const int pix   = base + lp;

    const float mx = sm_motion[lp * 2 + 0];
    const float my = sm_motion[lp * 2 + 1];

    // Normalized [-1,1] -> pixel coords (align_corners=False convention).
    const float gx  = (mx + 1.0f) * (WW * 0.5f) - 0.5f;
    const float gy  = (my + 1.0f) * (HH * 0.5f) - 0.5f;
    const float xwf = floorf(gx);
    const float ynf = floorf(gy);
    const float fx  = gx - xwf;
    const float fy  = gy - ynf;
    const float ex  = 1.0f - fx;
    const float sy  = 1.0f - fy;

    const int x0 = (int)xwf;
    const int y0 = (int)ynf;
    const int x1 = x0 + 1;
    const int y1 = y0 + 1;

    const bool bx0 = (x0 > -1) && (x0 < WW);
    const bool bx1 = (x1 > -1) && (x1 < WW);
    const bool by0 = (y0 > -1) && (y0 < HH);
    const bool by1 = (y1 > -1) && (y1 < HH);

    const bool m00 = by0 && bx0;   // nw
    const bool m01 = by0 && bx1;   // ne
    const bool m10 = by1 && bx0;   // sw
    const bool m11 = by1 && bx1;   // se

    const float w00 = sy * ex * (m00 ? 1.0f : 0.0f);
    const float w01 = sy * fx * (m01 ? 1.0f : 0.0f);
    const float w10 = fy * ex * (m10 ? 1.0f : 0.0f);
    const float w11 = fy * fx * (m11 ? 1.0f : 0.0f);

    // Clamped (masked) gather indices; weight is 0 when masked so the
    // dummy load from element 0 contributes nothing (matches reference).
    const int i00 = m00 ? (y0 * WW + x0) * CC : 0;
    const int i01 = m01 ? (y0 * WW + x1) * CC : 0;
    const int i10 = m10 ? (y1 * WW + x0) * CC : 0;
    const int i11 = m11 ? (y1 * WW + x1) * CC : 0;

    const int co = chunk * 4;

    // 8 consecutive lanes cover one contiguous 128 B source row per corner:
    // fully coalesced b128 gathers served from L2 (source is L2-resident).
    const v4f v00 = *(const v4f*)(src + i00 + co);
    const v4f v01 = *(const v4f*)(src + i01 + co);
    const v4f v10 = *(const v4f*)(src + i10 + co);
    const v4f v11 = *(const v4f*)(src + i11 + co);

    v4f acc = v00 * w00;
    acc += v01 * w01;
    acc += v10 * w10;
    acc += v11 * w11;

    // Non-temporal b128 store: keep the 184 MB output stream from evicting
    // the 2 MB source image out of L2.
    __builtin_nontemporal_store(acc, (v4f*)(out + (size_t)pix * CC + co));
}

extern "C" void kernel_launch(void* const* d_in, const int* in_sizes, int n_in,
                              void* d_out, int out_size, void* d_ws, size_t ws_size,
                              hipStream_t stream) {
    const float* src     = (const float*)d_in[0];  // (1,128,128,32) f32
    const float* motions = (const float*)d_in[1];  // (8,11,128,128,2) f32
    float* out           = (float*)d_out;          // (8,11,128,128,32) f32

    const int blocks = NPIX / PIX_PER_BLK;         // 45,056 blocks
    deform_grid_sample_kernel<<<blocks, TPB, 0, stream>>>(src, motions, out);
}